// GATCornerNet_35880156790904
// MI455X (gfx1250) — compile-verified
//
#include <hip/hip_runtime.h>
#include <math.h>

#define F_IN     128
#define EDGE_DIM 8
#define HID      64
#define HEADS    4
#define NEG_SLOPE 0.2f

typedef float v2f __attribute__((ext_vector_type(2)));
typedef float v8f __attribute__((ext_vector_type(8)));

// ---------- order-preserving float <-> uint encoding for atomic max ----------
__device__ __forceinline__ unsigned enc_f32(float f) {
  unsigned u = __float_as_uint(f);
  return (u & 0x80000000u) ? ~u : (u | 0x80000000u);
}
__device__ __forceinline__ float dec_f32(unsigned u) {
  return (u & 0x80000000u) ? __uint_as_float(u & 0x7fffffffu)
                           : __uint_as_float(~u);
}

// ---------------------------- utility kernels -------------------------------
__global__ void zero_kernel(float* __restrict__ p, long n) {
  long i = blockIdx.x * (long)blockDim.x + threadIdx.x;
  if (i < n) p[i] = 0.f;
}

// deg[d] += 1 ; ea_sum[d][:] += edge_attr[e][:]   (original E edges only)
__global__ void deg_ea_kernel(const int* __restrict__ dst,
                              const float* __restrict__ eattr,
                              float* __restrict__ deg,
                              float* __restrict__ ea_sum, int E) {
  int e = blockIdx.x * blockDim.x + threadIdx.x;
  if (e >= E) return;
  int d = dst[e];
  atomicAdd(&deg[d], 1.f);
  const float* ep = eattr + (size_t)e * EDGE_DIM;
  float* op = ea_sum + (size_t)d * EDGE_DIM;
#pragma unroll
  for (int j = 0; j < EDGE_DIM; ++j) atomicAdd(&op[j], ep[j]);
}

// ea_mean = ea_sum / max(deg, 1)    (in place)
__global__ void ea_mean_kernel(float* __restrict__ ea, const float* __restrict__ deg, int N) {
  long i = blockIdx.x * (long)blockDim.x + threadIdx.x;
  if (i >= (long)N * EDGE_DIM) return;
  float dg = deg[i / EDGE_DIM];
  ea[i] = ea[i] / fmaxf(dg, 1.f);
}

// ----------------------- fp32 WMMA GEMM: Out = A @ W -------------------------
// A: [nrows, fin] row-major, W: [fin, fout] row-major, Out: [nrows, fout].
// One wave computes a 16x16 output tile via V_WMMA_F32_16X16X4_F32 (K-step 4).
// nrows % 16 == 0, fin % 4 == 0, fout % 16 == 0 (all true for this model).
__global__ __launch_bounds__(256) void wmma_gemm_f32(
    const float* __restrict__ A, const float* __restrict__ W,
    float* __restrict__ Out, int nrows, int fin, int fout) {
  const int lane = threadIdx.x & 31;
  const int wave = threadIdx.x >> 5;
  const int tilesCol = fout >> 4;
  const int tilesTotal = (nrows >> 4) * tilesCol;
  const int tile = blockIdx.x * 8 + wave;       // wave-uniform guard
  if (tile >= tilesTotal) return;
  const int trow = tile / tilesCol;
  const int tcol = tile - trow * tilesCol;
  const int m0 = trow << 4;
  const int n0 = tcol << 4;
  const int half = lane >> 4;                   // 0: lanes 0-15, 1: lanes 16-31
  const int lmod = lane & 15;
  const size_t aRow = (size_t)(m0 + lmod) * fin;
  const int nB = n0 + lmod;

  v8f acc = (v8f){0.f, 0.f, 0.f, 0.f, 0.f, 0.f, 0.f, 0.f};
  for (int k = 0; k < fin; k += 4) {
    const int kk = k + 2 * half;
    // A 16x4: lane(0-15)=M rows, VGPR0 holds K = 2*half, VGPR1 K = 2*half+1
    v2f a;
    a.x = A[aRow + kk];
    a.y = A[aRow + kk + 1];
    // B 4x16: VGPR v holds row K = v + 2*half, cols striped over lanes
    v2f b;
    b.x = W[(size_t)kk * fout + nB];
    b.y = W[(size_t)(kk + 1) * fout + nB];
    acc = __builtin_amdgcn_wmma_f32_16x16x4_f32(
        /*neg_a=*/false, a, /*neg_b=*/false, b,
        /*c_mod=*/(short)0, acc, /*reuse_a=*/false, /*reuse_b=*/false);
  }
  // C/D 16x16: VGPR v -> row m0 + v + 8*half, col nB
  float* op = Out + (size_t)(m0 + 8 * half) * fout + nB;
#pragma unroll
  for (int v = 0; v < 8; ++v) op[(size_t)v * fout] = acc[v];
}

// ------------------- per-node attention coefficients ------------------------
// asrc[n,h] = <h[n,h,:], a_s[h,:]> ; adst likewise
__global__ void node_alpha_kernel(const float* __restrict__ hfeat,
                                  const float* __restrict__ a_s,
                                  const float* __restrict__ a_d,
                                  float* __restrict__ asrc,
                                  float* __restrict__ adst,
                                  int N, int H, int O) {
  long i = blockIdx.x * (long)blockDim.x + threadIdx.x;
  if (i >= (long)N * H) return;
  int h = (int)(i % H);
  long n = i / H;
  const float* hp = hfeat + (size_t)n * H * O + (size_t)h * O;
  const float* sp = a_s + (size_t)h * O;
  const float* dp = a_d + (size_t)h * O;
  float s1 = 0.f, s2 = 0.f;
  for (int o = 0; o < O; ++o) {
    float v = hp[o];
    s1 += v * sp[o];
    s2 += v * dp[o];
  }
  asrc[i] = s1;
  adst[i] = s2;
}

// vE[j,h] = sum_o We[j, h*O+o] * a_e[h,o]   (collapses the edge GEMM)
__global__ void ve_kernel(const float* __restrict__ We,
                          const float* __restrict__ a_e,
                          float* __restrict__ vE, int H, int O) {
  int idx = threadIdx.x;
  if (idx >= EDGE_DIM * H) return;
  int j = idx / H, h = idx % H;
  const float* wp = We + (size_t)j * H * O + (size_t)h * O;
  const float* ap = a_e + (size_t)h * O;
  float s = 0.f;
  for (int o = 0; o < O; ++o) s += wp[o] * ap[o];
  vE[idx] = s;   // stored [j*H + h]
}

// pass 1: e_val = leaky_relu(asrc[src]+adst[dst]+alpha_edge); atomic max per dst
__global__ void edge_score_kernel(const int* __restrict__ src,
                                  const int* __restrict__ dst,
                                  const float* __restrict__ eattr,
                                  const float* __restrict__ eam,
                                  const float* __restrict__ asrc,
                                  const float* __restrict__ adst,
                                  const float* __restrict__ vE,
                                  float* __restrict__ evalb,
                                  unsigned* __restrict__ mEnc,
                                  int E, int EF, int H) {
  long i = blockIdx.x * (long)blockDim.x + threadIdx.x;
  if (i >= (long)EF * H) return;
  int h = (int)(i % H);
  long e = i / H;
  int s, d;
  const float* ea;
  if (e < E) { s = src[e]; d = dst[e]; ea = eattr + (size_t)e * EDGE_DIM; }
  else       { s = d = (int)(e - E);   ea = eam + (size_t)(e - E) * EDGE_DIM; }
  float aedge = 0.f;
#pragma unroll
  for (int j = 0; j < EDGE_DIM; ++j) aedge += ea[j] * vE[j * H + h];
  float v = asrc[(size_t)s * H + h] + adst[(size_t)d * H + h] + aedge;
  v = (v > 0.f) ? v : NEG_SLOPE * v;
  evalb[i] = v;
  atomicMax(&mEnc[(size_t)d * H + h], enc_f32(v));
}

// pass 2: p = exp(e - m[dst]); sSum[dst] += p; overwrite evalb with p
__global__ void edge_softmax_kernel(const int* __restrict__ dst,
                                    float* __restrict__ evalb,
                                    const unsigned* __restrict__ mEnc,
                                    float* __restrict__ sSum,
                                    int E, int EF, int H) {
  long i = blockIdx.x * (long)blockDim.x + threadIdx.x;
  if (i >= (long)EF * H) return;
  int h = (int)(i % H);
  long e = i / H;
  int d = (e < E) ? dst[e] : (int)(e - E);
  float m = dec_f32(mEnc[(size_t)d * H + h]);
  float p = expf(evalb[i] - m);
  evalb[i] = p;
  atomicAdd(&sSum[(size_t)d * H + h], p);
}

// pass 3: agg[dst] += h[src] * (p / sSum[dst]); 4 features per thread
__global__ void edge_aggregate_kernel(const int* __restrict__ src,
                                      const int* __restrict__ dst,
                                      const float* __restrict__ hfeat,
                                      const float* __restrict__ pbuf,
                                      const float* __restrict__ sSum,
                                      float* __restrict__ agg,
                                      int E, int EF, int H, int O) {
  const int HO = H * O;
  const int quads = HO >> 2;
  long i = blockIdx.x * (long)blockDim.x + threadIdx.x;
  if (i >= (long)EF * quads) return;
  long e = i / quads;
  int q = (int)(i - e * quads);
  int c = q << 2;
  int h = c / O;
  int s, d;
  if (e < E) { s = src[e]; d = dst[e]; }
  else       { s = d = (int)(e - E); }
  float alpha = pbuf[(size_t)e * H + h] / sSum[(size_t)d * H + h];
  const float4 hv = *(const float4*)(hfeat + (size_t)s * HO + c);
  float* ap = agg + (size_t)d * HO + c;
  atomicAdd(ap + 0, hv.x * alpha);
  atomicAdd(ap + 1, hv.y * alpha);
  atomicAdd(ap + 2, hv.z * alpha);
  atomicAdd(ap + 3, hv.w * alpha);
}

// epilogue: out = agg + b ; mode 1: elu ; mode 2: elu + residual
__global__ void node_epilogue_kernel(const float* __restrict__ agg,
                                     const float* __restrict__ bias,
                                     const float* __restrict__ res,
                                     float* __restrict__ out,
                                     int N, int HO, int mode) {
  long i = blockIdx.x * (long)blockDim.x + threadIdx.x;
  if (i >= (long)N * HO) return;
  int c = (int)(i % HO);
  float v = agg[i] + bias[c];
  if (mode >= 1) v = (v > 0.f) ? v : (expf(v) - 1.f);  // ELU
  if (mode == 2) v += res[i];
  out[i] = v;
}

__global__ void bias_relu_kernel(const float* __restrict__ in,
                                 const float* __restrict__ b,
                                 float* __restrict__ out, long n, int C) {
  long i = blockIdx.x * (long)blockDim.x + threadIdx.x;
  if (i >= n) return;
  float v = in[i] + b[i % C];
  out[i] = v > 0.f ? v : 0.f;
}

// logits[n] = hidden[n,:] @ Wh2 + bh2
__global__ void head2_kernel(const float* __restrict__ hidden,
                             const float* __restrict__ Wh2,
                             const float* __restrict__ bh2,
                             float* __restrict__ out, int N) {
  int n = blockIdx.x * blockDim.x + threadIdx.x;
  if (n >= N) return;
  const float* hp = hidden + (size_t)n * 32;
  float s = bh2[0];
#pragma unroll
  for (int j = 0; j < 32; ++j) s += hp[j] * Wh2[j];
  out[n] = s;
}

// ------------------------------- host driver --------------------------------
static inline int cdiv(long n, int b) { return (int)((n + b - 1) / b); }

static void run_gat_layer(const float* hin, int fin, int H, int O,
                          const float* W, const float* We, const float* a_s,
                          const float* a_d, const float* a_e, const float* bias,
                          const float* residual, int mode, float* hout,
                          float* gemmH, float* asrc, float* adst,
                          unsigned* mEnc, float* sSum, float* agg,
                          float* evalb, float* vE,
                          const int* srcIdx, const int* dstIdx,
                          const float* eattr, const float* eam,
                          int N, int E, int EF, hipStream_t stream) {
  const int HO = H * O;
  // dense transform on WMMA
  int tiles = (N / 16) * (HO / 16);
  wmma_gemm_f32<<<cdiv(tiles, 8), 256, 0, stream>>>(hin, W, gemmH, N, fin, HO);
  node_alpha_kernel<<<cdiv((long)N * H, 256), 256, 0, stream>>>(
      gemmH, a_s, a_d, asrc, adst, N, H, O);
  ve_kernel<<<1, 64, 0, stream>>>(We, a_e, vE, H, O);
  // reset softmax state + accumulator
  zero_kernel<<<cdiv((long)N * H, 256), 256, 0, stream>>>((float*)mEnc, (long)N * H);
  zero_kernel<<<cdiv((long)N * H, 256), 256, 0, stream>>>(sSum, (long)N * H);
  zero_kernel<<<cdiv((long)N * HO, 256), 256, 0, stream>>>(agg, (long)N * HO);
  // edge passes
  edge_score_kernel<<<cdiv((long)EF * H, 256), 256, 0, stream>>>(
      srcIdx, dstIdx, eattr, eam, asrc, adst, vE, evalb, mEnc, E, EF, H);
  edge_softmax_kernel<<<cdiv((long)EF * H, 256), 256, 0, stream>>>(
      dstIdx, evalb, mEnc, sSum, E, EF, H);
  edge_aggregate_kernel<<<cdiv((long)EF * (HO / 4), 256), 256, 0, stream>>>(
      srcIdx, dstIdx, gemmH, evalb, sSum, agg, E, EF, H, O);
  node_epilogue_kernel<<<cdiv((long)N * HO, 256), 256, 0, stream>>>(
      agg, bias, residual, hout, N, HO, mode);
}

extern "C" void kernel_launch(void* const* d_in, const int* in_sizes, int n_in,
                              void* d_out, int out_size, void* d_ws, size_t ws_size,
                              hipStream_t stream) {
  // setup_inputs() insertion order: x, edge_index, edge_attr, then params dict
  const float* x     = (const float*)d_in[0];
  const int*   ei    = (const int*)d_in[1];
  const float* eattr = (const float*)d_in[2];
  const float* W0  = (const float*)d_in[3];
  const float* We0 = (const float*)d_in[4];
  const float* as0 = (const float*)d_in[5];
  const float* ad0 = (const float*)d_in[6];
  const float* ae0 = (const float*)d_in[7];
  const float* b0  = (const float*)d_in[8];
  const float* W1  = (const float*)d_in[9];
  const float* We1 = (const float*)d_in[10];
  const float* as1 = (const float*)d_in[11];
  const float* ad1 = (const float*)d_in[12];
  const float* ae1 = (const float*)d_in[13];
  const float* b1  = (const float*)d_in[14];
  const float* W2  = (const float*)d_in[15];
  const float* We2 = (const float*)d_in[16];
  const float* as2 = (const float*)d_in[17];
  const float* ad2 = (const float*)d_in[18];
  const float* ae2 = (const float*)d_in[19];
  const float* b2  = (const float*)d_in[20];
  const float* Wh1 = (const float*)d_in[21];
  const float* bh1 = (const float*)d_in[22];
  const float* Wh2 = (const float*)d_in[23];
  const float* bh2 = (const float*)d_in[24];

  const int N  = in_sizes[0] / F_IN;
  const int E  = in_sizes[2] / EDGE_DIM;
  const int EF = E + N;                 // + self loops
  const int* srcIdx = ei;               // edge_index row 0
  const int* dstIdx = ei + E;           // edge_index row 1
  const int HO = HEADS * HID;           // 256

  // ---- workspace layout (floats), all regions 16B aligned (N % 4 == 0) ----
  float* wsf   = (float*)d_ws;
  float* deg   = wsf;                         // N
  float* eam   = deg + N;                     // N*8
  float* h0    = eam + (size_t)N * 8;         // N*256
  float* h1    = h0 + (size_t)N * HO;         // N*256
  float* gemmH = h1 + (size_t)N * HO;         // N*256
  float* asrc  = gemmH + (size_t)N * HO;      // N*4
  float* adst  = asrc + (size_t)N * HEADS;    // N*4
  unsigned* mEnc = (unsigned*)(adst + (size_t)N * HEADS);  // N*4
  float* sSum  = (float*)mEnc + (size_t)N * HEADS;         // N*4
  float* agg   = sSum + (size_t)N * HEADS;    // N*256
  float* evalb = agg + (size_t)N * HO;        // EF*4
  float* vE    = evalb + (size_t)EF * HEADS;  // 64

  // ---- self-loop edge_attr = mean of incoming edge attrs ----
  zero_kernel<<<cdiv((long)N * 9, 256), 256, 0, stream>>>(deg, (long)N * 9);
  deg_ea_kernel<<<cdiv(E, 256), 256, 0, stream>>>(dstIdx, eattr, deg, eam, E);
  ea_mean_kernel<<<cdiv((long)N * 8, 256), 256, 0, stream>>>(eam, deg, N);

  // ---- layer 0: x[128] -> 4x64, ELU ----
  run_gat_layer(x, F_IN, HEADS, HID, W0, We0, as0, ad0, ae0, b0,
                nullptr, /*mode=*/1, h0,
                gemmH, asrc, adst, mEnc, sSum, agg, evalb, vE,
                srcIdx, dstIdx, eattr, eam, N, E, EF, stream);
  // ---- layer 1: 256 -> 4x64, ELU + residual ----
  run_gat_layer(h0, HO, HEADS, HID, W1, We1, as1, ad1, ae1, b1,
                h0, /*mode=*/2, h1,
                gemmH, asrc, adst, mEnc, sSum, agg, evalb, vE,
                srcIdx, dstIdx, eattr, eam, N, E, EF, stream);
  // ---- layer 2: 256 -> 1x64, no activation (h0 reused as output) ----
  run_gat_layer(h1, HO, 1, HID, W2, We2, as2, ad2, ae2, b2,
                nullptr, /*mode=*/0, h0,
                gemmH, asrc, adst, mEnc, sSum, agg, evalb, vE,
                srcIdx, dstIdx, eattr, eam, N, E, EF, stream);

  // ---- head MLP: relu(h2 @ Wh1 + bh1) @ Wh2 + bh2 ----
  int tiles = (N / 16) * (32 / 16);
  wmma_gemm_f32<<<cdiv(tiles, 8), 256, 0, stream>>>(h0, Wh1, gemmH, N, HID, 32);
  bias_relu_kernel<<<cdiv((long)N * 32, 256), 256, 0, stream>>>(
      gemmH, bh1, h1, (long)N * 32, 32);
  head2_kernel<<<cdiv(N, 256), 256, 0, stream>>>(h1, Wh2, bh2, (float*)d_out, N);
}